// InvariantPointAttention_86844238725637
// MI455X (gfx1250) — compile-verified
//
// Invariant Point Attention on MI455X (gfx1250): all matrix math via
// V_WMMA_F32_16X16X4_F32 (fp32 in/out). Workload is HBM/L2-bound on the
// 151MB zij tensor (2 passes ~= 300MB -> ~13us roofline), so fp32 WMMA is
// the right precision and B128 (float4) staging loads maximize bytes/instr.
#include <hip/hip_runtime.h>
#include <math.h>

typedef __attribute__((ext_vector_type(2))) float v2f;
typedef __attribute__((ext_vector_type(4))) float f4;
typedef __attribute__((ext_vector_type(8))) float v8f;

namespace {
constexpr int Bc = 2, Lc = 384, Cc = 128, Hc = 8, DKc = 16, NQPc = 4, NPVc = 8;
constexpr float WCc = 0.2357022603955158f;   // sqrt(2/(9*NQP))
constexpr float WLc = 0.5773502691896258f;   // sqrt(1/3)
constexpr int FEAT = Hc * Cc + Hc * DKc + Hc * NPVc * 3 + Hc * NPVc;  // 1408
}

__device__ __forceinline__ float softplusf(float x) {
  return (x > 0.f) ? (x + log1pf(expf(-x))) : log1pf(expf(x));
}

// ---------------------------------------------------------------------------
// Generic fp32 WMMA GEMM: C[M,N] = A[M,K] * B[K,N]  (row-major, ld* strides).
// Block = 128 threads (4 waves). Block tile: 16(M) x 128(N), K staged in LDS
// by 32 via float4 (global_load_b128) cooperative loads. Each wave owns two
// 16x16 output tiles and reuses its A fragment across both WMMAs.
// Batch (grid.z): Abase = A + (z/zdiv)*sA + (z%zdiv)*sA2 ; B,C linear in z.
// MODE 0: plain   MODE 1: += bias e0[n]
// MODE 2: scores epilogue: WL*(acc + bij - g*WC/2*(sqq[i]+sqk[j]))
// TRANSB: B supplied as [N,K] row-major (used for ka in the scores GEMM).
// Requires: K % 32 == 0, N % 4 == 0, lda/ldb % 4 == 0 (true for all uses).
// ---------------------------------------------------------------------------
template <int MODE, bool TRANSB>
__global__ __launch_bounds__(128) void k_gemm(
    const float* __restrict__ A, const float* __restrict__ Bm,
    float* __restrict__ Cm, int M, int N, int K, int lda, int ldb, int ldc,
    long long sA, long long sA2, int zdiv, long long sB, long long sC,
    int Mvalid, const float* __restrict__ e0, const float* __restrict__ e1,
    const float* __restrict__ e2, const float* __restrict__ e3) {
  __shared__ float As[16][36];
  __shared__ float Bs[32][132];

  const int z = blockIdx.z;
  const float* Ag = A + (long long)(z / zdiv) * sA + (long long)(z % zdiv) * sA2;
  const float* Bg = Bm + (long long)z * sB;
  float* Cg = Cm + (long long)z * sC;

  const int m0 = blockIdx.x * 16;
  const int n0 = blockIdx.y * 128;
  const int tid = threadIdx.x;
  const int wave = tid >> 5;
  const int lane = tid & 31;
  const int hf = lane >> 4;   // 0: lanes 0-15, 1: lanes 16-31
  const int l16 = lane & 15;
  const int nw = wave * 32;

  v8f acc0 = {};
  v8f acc1 = {};

  for (int kb = 0; kb < K; kb += 32) {
    // Stage A tile 16x32: one float4 per thread (coalesced B128 loads)
    {
      int r = tid >> 3, c4 = (tid & 7) * 4;
      f4 av = {0.f, 0.f, 0.f, 0.f};
      if (r < Mvalid && (m0 + r) < M)
        av = *(const f4*)(Ag + (long long)(m0 + r) * lda + kb + c4);
      *(f4*)&As[r][c4] = av;
    }
    // Stage B tile 32x128: eight float4 per thread
    if (!TRANSB) {
#pragma unroll
      for (int it = 0; it < 8; ++it) {
        int u = tid + it * 128;
        int r = u >> 5, c4 = (u & 31) * 4;
        f4 bv = {0.f, 0.f, 0.f, 0.f};
        if ((n0 + c4) < N)  // N%4==0 -> whole vector in bounds
          bv = *(const f4*)(Bg + (long long)(kb + r) * ldb + n0 + c4);
        *(f4*)&Bs[r][c4] = bv;
      }
    } else {
#pragma unroll
      for (int it = 0; it < 8; ++it) {
        int u = tid + it * 128;
        int c = u >> 3, r4 = (u & 7) * 4;  // vectorized along contiguous K
        f4 bv = {0.f, 0.f, 0.f, 0.f};
        if ((n0 + c) < N)
          bv = *(const f4*)(Bg + (long long)(n0 + c) * ldb + kb + r4);
        Bs[r4 + 0][c] = bv.x;
        Bs[r4 + 1][c] = bv.y;
        Bs[r4 + 2][c] = bv.z;
        Bs[r4 + 3][c] = bv.w;
      }
    }
    __syncthreads();

#pragma unroll
    for (int kk = 0; kk < 32; kk += 4) {
      // A 16x4 fragment: lanes 0-15 hold K=kk,kk+1; lanes 16-31 hold kk+2,kk+3
      v2f a, b0, b1;
      a.x = As[l16][kk + 2 * hf];
      a.y = As[l16][kk + 2 * hf + 1];
      // B 4x16 fragment: v0 = rows kk / kk+2 across lane halves, v1 = kk+1 / kk+3
      b0.x = Bs[kk + 2 * hf][nw + l16];
      b0.y = Bs[kk + 2 * hf + 1][nw + l16];
      b1.x = Bs[kk + 2 * hf][nw + 16 + l16];
      b1.y = Bs[kk + 2 * hf + 1][nw + 16 + l16];
      acc0 = __builtin_amdgcn_wmma_f32_16x16x4_f32(false, a, false, b0,
                                                   (short)0, acc0, false, false);
      acc1 = __builtin_amdgcn_wmma_f32_16x16x4_f32(false, a, false, b1,
                                                   (short)0, acc1, false, false);
    }
    __syncthreads();
  }

  float gw = 0.f;
  if (MODE == 2) {
    int h_ = z % Hc;
    gw = 0.5f * WCc * softplusf(e3[h_]);
  }

#pragma unroll
  for (int r = 0; r < 8; ++r) {
    int mrow = r + 8 * hf;  // C layout: VGPR r -> M=r (lanes 0-15) / M=r+8
    int m = m0 + mrow;
    if (mrow >= Mvalid || m >= M) continue;
#pragma unroll
    for (int t = 0; t < 2; ++t) {
      int n = n0 + nw + t * 16 + l16;
      if (n >= N) continue;
      float val = (t == 0) ? acc0[r] : acc1[r];
      if (MODE == 1) {
        val += e0[n];
      } else if (MODE == 2) {
        int b_ = z / Hc, h_ = z % Hc;
        float bijv = e0[((long long)(b_ * Lc + m) * Lc + n) * Hc + h_];
        val = WLc * (val + bijv -
                     gw * (e1[(long long)z * Lc + m] + e2[(long long)z * Lc + n]));
      }
      Cg[(long long)m * ldc + n] = val;
    }
  }
}

// ---------------------------------------------------------------------------
// Rigid transforms + operand packing. One thread per (b,i,h).
// qa[z,i,0:16]=0.25*q, qa[z,i,16:28]=g*WC*(R qp + t), rest 0 ; ka analogous
// (unscaled). vaug[z,j,0:16]=v, [16:40]=(R vp + t), [40:48)=0. sqq/sqk.
// ---------------------------------------------------------------------------
__global__ __launch_bounds__(256) void k_prep(
    const float* __restrict__ q, const float* __restrict__ k,
    const float* __restrict__ v, const float* __restrict__ qp,
    const float* __restrict__ kp, const float* __restrict__ vp,
    const float* __restrict__ Ri, const float* __restrict__ Ti,
    const float* __restrict__ gamah, float* __restrict__ qa,
    float* __restrict__ ka, float* __restrict__ vaug, float* __restrict__ sqq,
    float* __restrict__ sqk) {
  int idx = blockIdx.x * blockDim.x + threadIdx.x;
  if (idx >= Bc * Lc * Hc) return;
  int h = idx % Hc;
  int bi = idx / Hc;
  int b = bi / Lc, i = bi % Lc;
  int z = b * Hc + h;

  float R[3][3], t[3];
#pragma unroll
  for (int r = 0; r < 3; ++r)
#pragma unroll
    for (int c = 0; c < 3; ++c) R[r][c] = Ri[bi * 9 + r * 3 + c];
#pragma unroll
  for (int x = 0; x < 3; ++x) t[x] = Ti[bi * 3 + x];

  float gwc = WCc * softplusf(gamah[h]);

  float* qrow = qa + ((long long)z * Lc + i) * 32;
  float* krow = ka + ((long long)z * Lc + i) * 32;
  float* vrow = vaug + ((long long)z * Lc + i) * 48;

#pragma unroll
  for (int d = 0; d < DKc; ++d) {
    qrow[d] = 0.25f * q[(long long)bi * 128 + h * 16 + d];
    krow[d] = k[(long long)bi * 128 + h * 16 + d];
    vrow[d] = v[(long long)bi * 128 + h * 16 + d];
  }
  float sq = 0.f, sk = 0.f;
#pragma unroll
  for (int p = 0; p < NQPc; ++p) {
    const float* qs = qp + (long long)bi * 96 + h * 12 + p * 3;
    const float* ks = kp + (long long)bi * 96 + h * 12 + p * 3;
#pragma unroll
    for (int m = 0; m < 3; ++m) {
      float qt = R[m][0] * qs[0] + R[m][1] * qs[1] + R[m][2] * qs[2] + t[m];
      float kt = R[m][0] * ks[0] + R[m][1] * ks[1] + R[m][2] * ks[2] + t[m];
      sq += qt * qt;
      sk += kt * kt;
      qrow[16 + p * 3 + m] = gwc * qt;
      krow[16 + p * 3 + m] = kt;
    }
  }
#pragma unroll
  for (int d = 28; d < 32; ++d) { qrow[d] = 0.f; krow[d] = 0.f; }
#pragma unroll
  for (int p = 0; p < NPVc; ++p) {
    const float* vs = vp + (long long)bi * 192 + h * 24 + p * 3;
#pragma unroll
    for (int m = 0; m < 3; ++m)
      vrow[16 + p * 3 + m] =
          R[m][0] * vs[0] + R[m][1] * vs[1] + R[m][2] * vs[2] + t[m];
  }
#pragma unroll
  for (int d = 40; d < 48; ++d) vrow[d] = 0.f;
  sqq[(long long)z * Lc + i] = sq;
  sqk[(long long)z * Lc + i] = sk;
}

// Softmax over j (row length L=384). One wave32 per row, 8 rows per block.
__global__ __launch_bounds__(256) void k_softmax(float* __restrict__ sc) {
  int row = blockIdx.x * 8 + (threadIdx.x >> 5);
  int lane = threadIdx.x & 31;
  float* p = sc + (long long)row * Lc;
  float vals[12];
  float mx = -INFINITY;
#pragma unroll
  for (int e = 0; e < 12; ++e) {
    vals[e] = p[e * 32 + lane];
    mx = fmaxf(mx, vals[e]);
  }
#pragma unroll
  for (int o = 16; o >= 1; o >>= 1) mx = fmaxf(mx, __shfl_xor(mx, o, 32));
  float s = 0.f;
#pragma unroll
  for (int e = 0; e < 12; ++e) {
    vals[e] = expf(vals[e] - mx);
    s += vals[e];
  }
#pragma unroll
  for (int o = 16; o >= 1; o >>= 1) s += __shfl_xor(s, o, 32);
  float inv = 1.f / s;
#pragma unroll
  for (int e = 0; e < 12; ++e) p[e * 32 + lane] = vals[e] * inv;
}

// ohp = R^T (opt - t); write o_val / ohp / |ohp|-sums into feats.
__global__ __launch_bounds__(256) void k_finish(
    const float* __restrict__ ovalpt, const float* __restrict__ Ri,
    const float* __restrict__ Ti, float* __restrict__ feats) {
  int idx = blockIdx.x * blockDim.x + threadIdx.x;
  if (idx >= Bc * Lc * Hc) return;
  int h = idx % Hc;
  int bi = idx / Hc;
  int b = bi / Lc;
  int z = b * Hc + h;
  int i = bi % Lc;

  float R[3][3], t[3];
#pragma unroll
  for (int r = 0; r < 3; ++r)
#pragma unroll
    for (int c = 0; c < 3; ++c) R[r][c] = Ri[bi * 9 + r * 3 + c];
#pragma unroll
  for (int x = 0; x < 3; ++x) t[x] = Ti[bi * 3 + x];

  const float* orow = ovalpt + ((long long)z * Lc + i) * 48;
  float* frow = feats + (long long)bi * FEAT;

#pragma unroll
  for (int d = 0; d < DKc; ++d) frow[1024 + h * 16 + d] = orow[d];
#pragma unroll
  for (int p = 0; p < NPVc; ++p) {
    float d0 = orow[16 + p * 3 + 0] - t[0];
    float d1 = orow[16 + p * 3 + 1] - t[1];
    float d2 = orow[16 + p * 3 + 2] - t[2];
    float asum = 0.f;
#pragma unroll
    for (int m = 0; m < 3; ++m) {
      float oh = R[0][m] * d0 + R[1][m] * d1 + R[2][m] * d2;  // R^T
      frow[1152 + h * 24 + p * 3 + m] = oh;
      asum += fabsf(oh);
    }
    frow[1344 + h * 8 + p] = asum;
  }
}

// ---------------------------------------------------------------------------
static void launch_gemm(hipStream_t s, int mode, bool transb, const float* A,
                        const float* Bm, float* C, int M, int N, int K, int lda,
                        int ldb, int ldc, long long sA, long long sA2, int zdiv,
                        long long sB, long long sC, int Mvalid, int nz,
                        const float* e0 = nullptr, const float* e1 = nullptr,
                        const float* e2 = nullptr, const float* e3 = nullptr) {
  dim3 g((M + 15) / 16, (N + 127) / 128, nz);
  dim3 blk(128, 1, 1);
  if (mode == 0 && !transb)
    k_gemm<0, false><<<g, blk, 0, s>>>(A, Bm, C, M, N, K, lda, ldb, ldc, sA,
                                       sA2, zdiv, sB, sC, Mvalid, e0, e1, e2, e3);
  else if (mode == 1)
    k_gemm<1, false><<<g, blk, 0, s>>>(A, Bm, C, M, N, K, lda, ldb, ldc, sA,
                                       sA2, zdiv, sB, sC, Mvalid, e0, e1, e2, e3);
  else
    k_gemm<2, true><<<g, blk, 0, s>>>(A, Bm, C, M, N, K, lda, ldb, ldc, sA,
                                      sA2, zdiv, sB, sC, Mvalid, e0, e1, e2, e3);
}

extern "C" void kernel_launch(void* const* d_in, const int* in_sizes, int n_in,
                              void* d_out, int out_size, void* d_ws,
                              size_t ws_size, hipStream_t stream) {
  const float* si = (const float*)d_in[0];
  const float* zij = (const float*)d_in[1];
  const float* Ri = (const float*)d_in[2];
  const float* Ti = (const float*)d_in[3];
  const float* WQh = (const float*)d_in[4];
  const float* WKh = (const float*)d_in[5];
  const float* WVh = (const float*)d_in[6];
  const float* WQhp = (const float*)d_in[7];
  const float* WKhp = (const float*)d_in[8];
  const float* WVhp = (const float*)d_in[9];
  const float* Wb = (const float*)d_in[10];
  const float* WOw = (const float*)d_in[11];
  const float* WOb = (const float*)d_in[12];
  const float* gamah = (const float*)d_in[13];
  float* out = (float*)d_out;
  float* ws = (float*)d_ws;

  const int BL = Bc * Lc;                   // 768
  const long long LL = (long long)Lc * Lc;  // 147456
  float* q = ws;                     // [768,128]
  float* k = q + 98304;
  float* v = k + 98304;
  float* qp = v + 98304;             // [768,96]
  float* kp = qp + 73728;
  float* vp = kp + 73728;            // [768,192]
  float* qa = vp + 147456;           // [B,H,L,32]
  float* ka = qa + 196608;
  float* vaug = ka + 196608;         // [B,H,L,48]
  float* sqq = vaug + 294912;        // [B,H,L]
  float* sqk = sqq + 6144;
  float* bij = sqk + 6144;           // [B,L,L,H]
  float* alpha = bij + 2359296;      // [B,H,L,L] scores -> softmax in place
  float* ovalpt = alpha + 2359296;   // [B,H,L,48]
  float* feats = ovalpt + 294912;    // [768,1408]

  // 1) scalar + point projections (si @ W*)
  launch_gemm(stream, 0, false, si, WQh, q, BL, 128, Cc, Cc, 128, 128, 0, 0, 1, 0, 0, 16, 1);
  launch_gemm(stream, 0, false, si, WKh, k, BL, 128, Cc, Cc, 128, 128, 0, 0, 1, 0, 0, 16, 1);
  launch_gemm(stream, 0, false, si, WVh, v, BL, 128, Cc, Cc, 128, 128, 0, 0, 1, 0, 0, 16, 1);
  launch_gemm(stream, 0, false, si, WQhp, qp, BL, 96, Cc, Cc, 96, 96, 0, 0, 1, 0, 0, 16, 1);
  launch_gemm(stream, 0, false, si, WKhp, kp, BL, 96, Cc, Cc, 96, 96, 0, 0, 1, 0, 0, 16, 1);
  launch_gemm(stream, 0, false, si, WVhp, vp, BL, 192, Cc, Cc, 192, 192, 0, 0, 1, 0, 0, 16, 1);

  // 2) pair bias: zij[B*L*L,128] @ Wb[128,8] -> bij  (first zij pass)
  launch_gemm(stream, 0, false, zij, Wb, bij, Bc * Lc * Lc, Hc, Cc, Cc, Hc, Hc,
              0, 0, 1, 0, 0, 16, 1);

  // 3) rigid transforms + packed WMMA operands
  k_prep<<<(Bc * Lc * Hc + 255) / 256, 256, 0, stream>>>(
      q, k, v, qp, kp, vp, Ri, Ti, gamah, qa, ka, vaug, sqq, sqk);

  // 4) fused scores: per (b,h) 384x384x32 GEMM with bias/point epilogue
  launch_gemm(stream, 2, true, qa, ka, alpha, Lc, Lc, 32, 32, 32, Lc,
              (long long)Lc * 32, 0, 1, (long long)Lc * 32, LL, 16, Bc * Hc,
              bij, sqq, sqk, gamah);

  // 5) softmax over j
  k_softmax<<<(Bc * Hc * Lc) / 8, 256, 0, stream>>>(alpha);

  // 6) o_val + opt: per (b,h) alpha[384,384] @ vaug[384,48]
  launch_gemm(stream, 0, false, alpha, vaug, ovalpt, Lc, 48, Lc, Lc, 48, 48,
              LL, 0, 1, (long long)Lc * 48, (long long)Lc * 48, 16, Bc * Hc);

  // 7) o_pair: per (b,i) alpha[b,:,i,:][8,384] @ zij[b,i][384,128] -> feats
  //    (second zij pass; zij fits in the 192MB L2 so this mostly hits L2)
  launch_gemm(stream, 0, false, alpha, zij, feats, 16, Cc, Lc, (int)LL, Cc,
              Cc, (long long)Hc * LL, (long long)Lc, Lc, (long long)Lc * Cc,
              (long long)FEAT, Hc, BL);

  // 8) inverse transform + abs-sum features
  k_finish<<<(Bc * Lc * Hc + 255) / 256, 256, 0, stream>>>(ovalpt, Ri, Ti, feats);

  // 9) output projection: feats[768,1408] @ WOw[1408,128] + WOb
  launch_gemm(stream, 1, false, feats, WOw, out, BL, Cc, FEAT, FEAT, Cc, Cc,
              0, 0, 1, 0, 0, 16, 1, WOb);
}